// MMD_7086696038964
// MI455X (gfx1250) — compile-verified
//
#include <hip/hip_runtime.h>
#include <hip/hip_bf16.h>

typedef __attribute__((ext_vector_type(2))) float v2f;
typedef __attribute__((ext_vector_type(8))) float v8f;

#define DIM  256   // embedding dimension
#define BM   128   // block tile rows
#define BN   128   // block tile cols
#define NT   (BN / 16)
#define KC   32    // K chunk staged in LDS
#define LSTR 36    // padded LDS row stride (floats): conflict-free b64 pattern

// ---------------------------------------------------------------------------
// Kernel 1: signed multiplicity weights from labels.
// w[r] = +count for src rows (label != 1), -count for tgt rows (label==1).
// Degenerate single-class case -> all-zero weights -> MMD = 0 (matches ref).
// ---------------------------------------------------------------------------
__global__ __launch_bounds__(256) void mmd_prep_weights(const int* __restrict__ labels,
                                                        float* __restrict__ w, int n) {
  if (blockIdx.x != 0 || threadIdx.x != 0) return;
  int nt = 0;
  for (int r = 0; r < n; ++r) nt += (labels[r] == 1) ? 1 : 0;
  int ns = n - nt;
  if (ns == 0 || nt == 0) {
    for (int r = 0; r < n; ++r) w[r] = 0.0f;
    return;
  }
  int cs = 0, ct = 0;
  for (int r = 0; r < n; ++r) {
    bool tgt = (labels[r] == 1);
    int p   = tgt ? ct++ : cs++;
    int den = tgt ? nt : ns;
    int c   = (n - 1 - p) / den + 1;          // multiplicity in [src|tgt]_rep
    w[r] = tgt ? -(float)c : (float)c;
  }
}

// ---------------------------------------------------------------------------
// Kernel 2: per-row squared L2 norms.
// ---------------------------------------------------------------------------
__global__ __launch_bounds__(256) void mmd_rowsq(const float* __restrict__ E,
                                                 float* __restrict__ sq, int n) {
  int r = blockIdx.x * 256 + threadIdx.x;
  if (r >= n) return;
  const float4* p = (const float4*)(E + (size_t)r * DIM);
  float s = 0.0f;
#pragma unroll 8
  for (int i = 0; i < DIM / 4; ++i) {
    float4 v = p[i];
    s += v.x * v.x + v.y * v.y + v.z * v.z + v.w * v.w;
  }
  sq[r] = s;
}

// ---------------------------------------------------------------------------
// Kernel 3: main WMMA kernel. Computes only lower-triangular blocks
// (by <= bx) of the symmetric Gram matrix with V_WMMA_F32_16X16X4_F32;
// off-diagonal block contributions are doubled. Gaussian kernel-sum epilogue
// weighted by w[m]*w[n]; one deterministic partial per block.
// ---------------------------------------------------------------------------
__global__ __launch_bounds__(256) void mmd_main(const float* __restrict__ E,
                                                const float* __restrict__ sq,
                                                const float* __restrict__ w,
                                                float* __restrict__ part) {
  const int bx = blockIdx.x, by = blockIdx.y;
  const int tid = threadIdx.x;

  // Symmetry: skip upper-triangular blocks; zero their partial slot so the
  // final reduction never reads poisoned workspace.
  if (by > bx) {
    if (tid == 0) part[by * gridDim.x + bx] = 0.0f;
    return;
  }

  __shared__ __align__(16) float Ash[BM * LSTR];
  __shared__ __align__(16) float Bsh[BN * LSTR];
  __shared__ float wsum[8];

  const int mBase = by * BM;
  const int nBase = bx * BN;
  const int wave  = tid >> 5;
  const int lane  = tid & 31;
  const int l16   = lane & 15;
  const int lhi   = lane >> 4;

  v8f zero = {};
  v8f acc[NT];
#pragma unroll
  for (int t = 0; t < NT; ++t) acc[t] = zero;

  for (int kc = 0; kc < DIM; kc += KC) {
    // Stage A (rows mBase..+127) and B (rows nBase..+127), KC columns, into LDS.
#pragma unroll
    for (int i = 0; i < (BM * KC / 4) / 256; ++i) {
      int linear = tid + i * 256;
      int row = linear >> 3;          // 8 float4 per row
      int c4  = (linear & 7) << 2;
      float4 va = *(const float4*)&E[(size_t)(mBase + row) * DIM + kc + c4];
      *(float4*)&Ash[row * LSTR + c4] = va;
      float4 vb = *(const float4*)&E[(size_t)(nBase + row) * DIM + kc + c4];
      *(float4*)&Bsh[row * LSTR + c4] = vb;
    }
    __syncthreads();

    // f32 16x16x4 fragments: lane L, vgpr g <-> E[row + L%16][k + 2*(L/16) + g]
    const float* aptr  = &Ash[(wave * 16 + l16) * LSTR + lhi * 2];
    const float* bbase = &Bsh[l16 * LSTR + lhi * 2];
#pragma unroll
    for (int k = 0; k < KC; k += 4) {
      v2f a = *(const v2f*)(aptr + k);
      // Batch all B-fragment loads into distinct registers so the compiler
      // can issue ds_loads back-to-back and wait with non-zero dscnt.
      v2f b[NT];
#pragma unroll
      for (int t = 0; t < NT; ++t)
        b[t] = *(const v2f*)(bbase + t * 16 * LSTR + k);
#pragma unroll
      for (int t = 0; t < NT; ++t)
        acc[t] = __builtin_amdgcn_wmma_f32_16x16x4_f32(
            /*neg_a=*/false, a, /*neg_b=*/false, b[t],
            /*c_mod=*/(short)0, acc[t], /*reuse_a=*/false, /*reuse_b=*/false);
    }
    __syncthreads();
  }

  // Gaussian kernel-sum epilogue. bandwidths: 1/16, 1/4, 1, 4 (+eps).
  const float inv0 = 1.0f / (0.0625f + 1e-5f);
  const float inv1 = 1.0f / (0.25f   + 1e-5f);
  const float inv2 = 1.0f / (1.0f    + 1e-5f);
  const float inv3 = 1.0f / (4.0f    + 1e-5f);

  // C/D layout: vgpr v, lanes 0-15 -> (M=v, N=lane); lanes 16-31 -> (M=v+8, N=lane-16)
  float wm[8], sm[8];
#pragma unroll
  for (int v = 0; v < 8; ++v) {
    int m = mBase + wave * 16 + lhi * 8 + v;
    wm[v] = w[m];
    sm[v] = sq[m];
  }

  float partial = 0.0f;
#pragma unroll
  for (int t = 0; t < NT; ++t) {
    int nIdx = nBase + t * 16 + l16;
    float wn = w[nIdx];
    float sn = sq[nIdx];
#pragma unroll
    for (int v = 0; v < 8; ++v) {
      float l2 = sm[v] + sn - 2.0f * acc[t][v];
      l2 = fmaxf(l2, 0.0f);
      float kv = __expf(-l2 * inv0) + __expf(-l2 * inv1) +
                 __expf(-l2 * inv2) + __expf(-l2 * inv3);
      partial += wm[v] * wn * kv;
    }
  }

  // Off-diagonal blocks stand in for their mirrored twin.
  partial *= (bx == by) ? 1.0f : 2.0f;

  // Deterministic wave32 reduction, then per-block LDS reduction.
#pragma unroll
  for (int off = 16; off > 0; off >>= 1)
    partial += __shfl_down(partial, off, 32);
  if (lane == 0) wsum[wave] = partial;
  __syncthreads();
  if (tid == 0) {
    float s = 0.0f;
#pragma unroll
    for (int i = 0; i < 8; ++i) s += wsum[i];
    part[by * gridDim.x + bx] = s;
  }
}

// ---------------------------------------------------------------------------
// Kernel 4: deterministic fixed-order reduction of block partials.
// ---------------------------------------------------------------------------
__global__ __launch_bounds__(256) void mmd_reduce(const float* __restrict__ part, int nPart,
                                                  float* __restrict__ out, float scale) {
  __shared__ float sh[256];
  float s = 0.0f;
  for (int i = threadIdx.x; i < nPart; i += 256) s += part[i];
  sh[threadIdx.x] = s;
  __syncthreads();
  for (int off = 128; off > 0; off >>= 1) {
    if ((int)threadIdx.x < off) sh[threadIdx.x] += sh[threadIdx.x + off];
    __syncthreads();
  }
  if (threadIdx.x == 0) out[0] = sh[0] * scale;
}

extern "C" void kernel_launch(void* const* d_in, const int* in_sizes, int n_in,
                              void* d_out, int out_size, void* d_ws, size_t ws_size,
                              hipStream_t stream) {
  const float* E      = (const float*)d_in[0];   // 4096 x 256 fp32
  const int*   labels = (const int*)d_in[1];     // 4096 ints
  float* out = (float*)d_out;

  int n = in_sizes[1];                // 4096 rows
  float* w    = (float*)d_ws;         // n floats
  float* sq   = w + n;                // n floats
  float* part = sq + n;               // (n/BM)*(n/BN) floats

  int gm = n / BM, gn = n / BN;

  mmd_prep_weights<<<1, 32, 0, stream>>>(labels, w, n);
  mmd_rowsq<<<(n + 255) / 256, 256, 0, stream>>>(E, sq, n);
  dim3 grid(gn, gm);
  mmd_main<<<grid, 256, 0, stream>>>(E, sq, w, part);
  float scale = 1.0f / ((float)n * (float)n);
  mmd_reduce<<<1, 256, 0, stream>>>(part, gm * gn, out, scale);
}